// GroupedConvFuse_18992345383203
// MI455X (gfx1250) — compile-verified
//
#include <hip/hip_runtime.h>

typedef __attribute__((ext_vector_type(2))) float v2f;
typedef __attribute__((ext_vector_type(8))) float v8f;

#define BATCH 8
#define NCLS  19
#define HW_SZ (512 * 512)   /* = 2^18 */

#if defined(__HIP_DEVICE_COMPILE__)
#if !__has_builtin(__builtin_amdgcn_wmma_f32_16x16x4_f32)
#error "builtin __builtin_amdgcn_wmma_f32_16x16x4_f32 not available"
#endif
#endif

// D = A(16x4 f32: weight[:,1:5]) x B(4x16 f32: lows for 16 pixels) + C
// C preloaded with side5*w0 + bias, so one WMMA completes 16 pixels x 16 classes.
__global__ __launch_bounds__(256) void grouped_conv_fuse_wmma(
    const float* __restrict__ side1,
    const float* __restrict__ side2,
    const float* __restrict__ side3,
    const float* __restrict__ side4,
    const float* __restrict__ side5,   // [B, 19, H, W]
    const float* __restrict__ weight,  // [19, 5]
    const float* __restrict__ bias,    // [19]
    float* __restrict__ out)           // [B, 19, H, W]
{
    const int lane = threadIdx.x & 31;
    const int half = lane >> 4;   // 0 = lanes 0-15 (K0/K1), 1 = lanes 16-31 (K2/K3)
    const int n    = lane & 15;   // pixel column within the 16-wide tile

    // ---- A matrices (weights), 16(M=class) x 4(K=side). ISA layout:
    // lanes 0-15: M=lane, VGPR0=K0, VGPR1=K1 ; lanes 16-31: M=lane-16, VGPR0=K2, VGPR1=K3
    const int kbase = 1 + 2 * half;          // weight column for this half's K pair
    v2f a1, a2;
    a1.x = weight[n * 5 + kbase];
    a1.y = weight[n * 5 + kbase + 1];
    const int cls2 = 16 + n;                 // tile-2 class for this lane's M row
    a2.x = (cls2 < NCLS) ? weight[cls2 * 5 + kbase]     : 0.0f;
    a2.y = (cls2 < NCLS) ? weight[cls2 * 5 + kbase + 1] : 0.0f;

    // Per-C/D-row class ids + w0 + bias (tile1: classes 0..15, tile2: 16..18)
    int   kt1[8];  float w0t1[8], bst1[8];
#pragma unroll
    for (int r = 0; r < 8; ++r) {
        const int k = r + half * 8;
        kt1[r] = k;  w0t1[r] = weight[k * 5];  bst1[r] = bias[k];
    }
    int   kt2[3];  float w0t2[3], bst2[3];
#pragma unroll
    for (int r = 0; r < 3; ++r) {
        int k = 16 + r + half * 8;
        if (k > NCLS - 1) k = NCLS - 1;      // clamp: lanes 16-31 rows are dead
        kt2[r] = k;  w0t2[r] = weight[k * 5];  bst2[r] = bias[k];
    }

    // Low-side pointers for this half's K rows of the B matrix
    const float* __restrict__ lowA = half ? side3 : side1;   // K0 / K2
    const float* __restrict__ lowB = half ? side4 : side2;   // K1 / K3

    const long nGroups     = (long)BATCH * HW_SZ / 16;       // 16-pixel tiles
    const int  wavesPerBlk = blockDim.x >> 5;
    const long nWaves      = (long)gridDim.x * wavesPerBlk;
    const long gw          = (long)blockIdx.x * wavesPerBlk + (threadIdx.x >> 5);

    for (long g = gw; g < nGroups; g += nWaves) {            // wave-uniform loop
        const long pixBase = g << 4;
        const int  b  = (int)(pixBase >> 18);                // HW_SZ = 2^18
        const int  p0 = (int)(pixBase & (HW_SZ - 1));

        const long lowOff = (long)b * HW_SZ + p0 + n;
        v2f bm;                                              // B: 4(K) x 16(N)
        bm.x = lowA[lowOff];
        bm.y = lowB[lowOff];

        const float* __restrict__ s5 = side5 + (long)b * NCLS * HW_SZ + p0 + n;
        float*       __restrict__ op = out   + (long)b * NCLS * HW_SZ + p0 + n;

        // C preload: side5*w0 + bias   (side5 is streamed once -> non-temporal)
        v8f c1, c2;
#pragma unroll
        for (int r = 0; r < 8; ++r) {
            const float s = __builtin_nontemporal_load(s5 + (long)kt1[r] * HW_SZ);
            c1[r] = s * w0t1[r] + bst1[r];
        }
#pragma unroll
        for (int r = 0; r < 3; ++r) {
            const float s = __builtin_nontemporal_load(s5 + (long)kt2[r] * HW_SZ);
            c2[r] = s * w0t2[r] + bst2[r];
        }
#pragma unroll
        for (int r = 3; r < 8; ++r) c2[r] = 0.0f;

        v8f d1 = __builtin_amdgcn_wmma_f32_16x16x4_f32(false, a1, false, bm,
                                                       (short)0, c1, false, false);
        v8f d2 = __builtin_amdgcn_wmma_f32_16x16x4_f32(false, a2, false, bm,
                                                       (short)0, c2, false, false);

        // Stores: output streamed once -> non-temporal
#pragma unroll
        for (int r = 0; r < 8; ++r)
            __builtin_nontemporal_store(d1[r], op + (long)kt1[r] * HW_SZ);
#pragma unroll
        for (int r = 0; r < 3; ++r)          // classes 16..18 live in lanes 0-15 only
            if (half == 0)
                __builtin_nontemporal_store(d2[r], op + (long)(16 + r) * HW_SZ);
    }
}

extern "C" void kernel_launch(void* const* d_in, const int* in_sizes, int n_in,
                              void* d_out, int out_size, void* d_ws, size_t ws_size,
                              hipStream_t stream) {
    const float* side1  = (const float*)d_in[0];
    const float* side2  = (const float*)d_in[1];
    const float* side3  = (const float*)d_in[2];
    const float* side4  = (const float*)d_in[3];
    const float* side5  = (const float*)d_in[4];
    const float* weight = (const float*)d_in[5];
    const float* bias   = (const float*)d_in[6];
    float*       out    = (float*)d_out;

    // 2048 blocks x 256 threads = 16384 waves; 131072 pixel-tiles -> 8 tiles/wave
    grouped_conv_fuse_wmma<<<dim3(2048), dim3(256), 0, stream>>>(
        side1, side2, side3, side4, side5, weight, bias, out);
}